// RoiPooling_8710193676768
// MI455X (gfx1250) — compile-verified
//
#include <hip/hip_runtime.h>

// ROI-align bilinear resize, MI455X (gfx1250, wave32).
// out[b, r, py, px, c] = bilerp of img[b, :, :, c] at roi-relative coords.
//
// Roofline: 25.7MB written + ~103MB tap reads (L2-cacheable: 64MB/image vs
// 192MB L2) against ~77 MFLOP -> ~1.7:1 bytes:flops, purely HBM-bound.
// No matmul structure (weights AND tap addresses vary per output pixel),
// so WMMA/TDM do not apply; the win is wave32 b128 coalescing + cache hints.
//
// Mapping: grid = (7, 512) = (py, b*128+r); block = 448 = 7 px-groups x 64
// channel-lanes. Each wave32 is one px-group half: 32 lanes x float4 = 512B
// contiguous per tap load and per store. ROI fetch is block-uniform -> SMEM.

typedef float v4f __attribute__((ext_vector_type(4)));

namespace {
constexpr int kPool = 7;
constexpr int kRois = 128;
constexpr int kB = 4, kH = 256, kW = 256, kC = 256;
}

__device__ __forceinline__ void axis_coords(int start, int size, int p,
                                            int& g0, int& g1, float& f) {
  const float sz = (float)size;
  float src = ((float)p + 0.5f) * sz / 7.0f - 0.5f;
  src = fminf(fmaxf(src, 0.0f), sz - 1.0f);
  const float fl = floorf(src);
  const int i0 = (int)fl;
  f = src - fl;
  const int i1 = (i0 + 1 < size) ? (i0 + 1) : (size - 1);
  g0 = start + i0;
  g1 = start + i1;
}

__global__ __launch_bounds__(448) void roi_bilinear_kernel(
    const float* __restrict__ img, const int* __restrict__ rois,
    float* __restrict__ out) {
  const int tid = threadIdx.x;
  const int px  = tid >> 6;        // 0..6, uniform per wave
  const int c4  = (tid & 63) << 2; // channel offset, contiguous per wave
  const int py  = blockIdx.x;      // 0..6
  const int br  = blockIdx.y;      // b*kRois + r, 0..511

  // roi = {x, y, w, h}; address is block-uniform -> scalar s_load_b128.
  const int4 roi = ((const int4*)rois)[br];

  int gx0, gx1, gy0, gy1;
  float fx, fy;
  axis_coords(roi.x, roi.z, px, gx0, gx1, fx);
  axis_coords(roi.y, roi.w, py, gy0, gy1, fy);

  // All element indices < 2^27: 32-bit offsets -> saddr + voffset forms.
  const float* imgb = img + (size_t)(br >> 7) * (kH * kW * kC);
  const float* r0 = imgb + gy0 * (kW * kC);
  const float* r1 = imgb + gy1 * (kW * kC);

  // Four independent 128-bit tap loads (one s_wait_loadcnt covers all).
  // Default RT temporal hint: taps are reused by neighboring output pixels
  // and the image stays L2-resident.
  const v4f v00 = *(const v4f*)(r0 + gx0 * kC + c4);
  const v4f v01 = *(const v4f*)(r0 + gx1 * kC + c4);
  const v4f v10 = *(const v4f*)(r1 + gx0 * kC + c4);
  const v4f v11 = *(const v4f*)(r1 + gx1 * kC + c4);

  const float ofx = 1.0f - fx;
  const float ofy = 1.0f - fy;

  v4f top, bot, o;
  top.x = v00.x * ofx + v01.x * fx;
  top.y = v00.y * ofx + v01.y * fx;
  top.z = v00.z * ofx + v01.z * fx;
  top.w = v00.w * ofx + v01.w * fx;
  bot.x = v10.x * ofx + v11.x * fx;
  bot.y = v10.y * ofx + v11.y * fx;
  bot.z = v10.z * ofx + v11.z * fx;
  bot.w = v10.w * ofx + v11.w * fx;
  o.x = top.x * ofy + bot.x * fy;
  o.y = top.y * ofy + bot.y * fy;
  o.z = top.z * ofy + bot.z * fy;
  o.w = top.w * ofy + bot.w * fy;

  // Write-once output: non-temporal b128 store (gfx1250 TH=NT hint) keeps
  // the 25.7MB result stream from evicting L2-resident image data.
  const int pix = (br * kPool + py) * kPool + px;
  float* dst = out + (size_t)pix * kC + c4;
  __builtin_nontemporal_store(o, (v4f*)dst);
}

extern "C" void kernel_launch(void* const* d_in, const int* in_sizes, int n_in,
                              void* d_out, int out_size, void* d_ws, size_t ws_size,
                              hipStream_t stream) {
  const float* img  = (const float*)d_in[0];
  const int*   rois = (const int*)d_in[1];
  float*       out  = (float*)d_out;
  (void)in_sizes; (void)n_in; (void)out_size; (void)d_ws; (void)ws_size;

  roi_bilinear_kernel<<<dim3(kPool, kB * kRois), dim3(448), 0, stream>>>(img, rois, out);
}